// TaskAlignedAssigner_21560735826416
// MI455X (gfx1250) — compile-verified
//
#include <hip/hip_runtime.h>
#include <stdint.h>

// ---------------- problem constants (from reference) ----------------
#define TOPK 13
#define NPRO 31
#define NALP 24
#define NADS 37
#define BS   32
#define NA   8400
#define NM   64
#define EPSF 1e-9f

#define NT    256            // 8 wave32s per block
#define CHUNK 1024           // anchors staged per LDS tile
#define NCAND (NT*TOPK)      // 3328 merge candidates per row

// ---------------- output layout (flat concat, floats) ----------------
constexpr size_t PBA      = (size_t)BS*NA;            // 268800
constexpr size_t OFF_TPRO = 0;
constexpr size_t OFF_TALP = OFF_TPRO + PBA;
constexpr size_t OFF_TADS = OFF_TALP + PBA;           // (5,BS,NA)
constexpr size_t OFF_TBB  = OFF_TADS + 5*PBA;         // (BS,NA,4)
constexpr size_t OFF_TCR  = OFF_TBB  + 4*PBA;         // (BS,NA,8)
constexpr size_t OFF_PRO  = OFF_TCR  + 8*PBA;         // (BS,NA,31)
constexpr size_t OFF_ALP  = OFF_PRO  + (size_t)NPRO*PBA;
constexpr size_t OFF_ADS  = OFF_ALP  + (size_t)NALP*PBA; // (5,BS,NA,37)
constexpr size_t OFF_FG   = OFF_ADS  + (size_t)5*NADS*PBA;

// Non-temporal store: 280 MB of write-once output > 192 MB L2; keep the
// L2 for the overlaps/align workspace that k3 re-reads.
__device__ __forceinline__ void st_nt(float* p, float v) {
  __builtin_nontemporal_store(v, p);
}

// ---------------- CDNA5 async global->LDS helpers ----------------
// gfx1250: GLOBAL_LOAD_ASYNC_TO_LDS_B128 (tracked by ASYNCcnt).
// vdst = VGPR holding LDS byte address, vaddr = 64-bit global address.
__device__ __forceinline__ unsigned lds_addr_of(const void* p) {
  // generic LDS pointer = {shared aperture, lds_addr[31:0]}; low 32 bits are
  // the LDS address the VFLAT async encoding wants.
  return (unsigned)(unsigned long long)p;
}
__device__ __forceinline__ void async_ld_b128(unsigned lds_addr, const void* gaddr) {
  asm volatile("global_load_async_to_lds_b128 %0, %1, off"
               :: "v"(lds_addr), "v"((unsigned long long)gaddr)
               : "memory");
}
__device__ __forceinline__ void wait_async0() {
  asm volatile("s_wait_asynccnt 0" ::: "memory");
}

// =====================================================================
// k0: zero posbits + per-row stats
// =====================================================================
__global__ __launch_bounds__(NT) void k0_init(unsigned long long* __restrict__ posbits,
                                              unsigned* __restrict__ pa,
                                              unsigned* __restrict__ po) {
  int i = blockIdx.x * NT + threadIdx.x;
  if (i < (int)PBA) posbits[i] = 0ull;
  if (i < BS*NM) { pa[i] = 0u; po[i] = 0u; }
}

// =====================================================================
// k1: per (b,m) row -> overlaps, align to ws + exact top-13 indices.
// Async-stages pd_bboxes / anc_points tiles into LDS.
// =====================================================================
__global__ __launch_bounds__(NT) void k1_metrics_topk(
    const float* __restrict__ pro,    // (BS,NA,NPRO)
    const float* __restrict__ pdbb,   // (BS,NA,4)
    const float* __restrict__ anc,    // (NA,2)
    const int*   __restrict__ gtpro,  // (BS,NM,1)
    const float* __restrict__ gtbb,   // (BS,NM,4)
    float* __restrict__ ov_ws,        // (BS,NM,NA)
    float* __restrict__ al_ws,        // (BS,NM,NA)
    int*   __restrict__ topk_ws)      // (BS,NM,TOPK)
{
  __shared__ __align__(16) float bb_tile[CHUNK*4];   // 16 KB
  __shared__ __align__(16) float an_tile[CHUNK*2];   //  8 KB
  __shared__ float cv[NCAND];                        // 13 KB
  __shared__ int   ci[NCAND];                        // 13 KB
  __shared__ float rv[NT];
  __shared__ int   ri[NT];

  const int row = blockIdx.x;        // b*NM + m
  const int b   = row / NM;
  const int tid = threadIdx.x;

  const float gx1 = gtbb[row*4+0], gy1 = gtbb[row*4+1];
  const float gx2 = gtbb[row*4+2], gy2 = gtbb[row*4+3];
  const int   lab = gtpro[row];
  const float ga  = fmaxf(gx2-gx1, 0.f) * fmaxf(gy2-gy1, 0.f);

  float tv[TOPK]; int tix[TOPK];
#pragma unroll
  for (int k = 0; k < TOPK; ++k) { tv[k] = -1.f; tix[k] = 0; }

  const float* bbB = pdbb + (size_t)b*NA*4;
  const float* prB = pro  + (size_t)b*NA*NPRO + lab;
  float* ovR = ov_ws + (size_t)row*NA;
  float* alR = al_ws + (size_t)row*NA;

  for (int a0 = 0; a0 < NA; a0 += CHUNK) {
    const int cnt = min(CHUNK, NA - a0);
    // --- async stage: pd_bboxes tile (cnt * 16B) and anc tile (cnt * 8B) ---
    for (int i = tid; i < cnt; i += NT)
      async_ld_b128(lds_addr_of(&bb_tile[i*4]), bbB + (size_t)(a0 + i)*4);
    for (int i = tid; i < (cnt >> 1); i += NT)
      async_ld_b128(lds_addr_of(&an_tile[i*4]), anc + (size_t)(a0 + i*2)*2);
    wait_async0();        // per-wave ASYNCcnt drain
    __syncthreads();      // cross-wave visibility of LDS tile

    for (int i = tid; i < cnt; i += NT) {
      const int a = a0 + i;
      const float px1 = bb_tile[i*4+0], py1 = bb_tile[i*4+1];
      const float px2 = bb_tile[i*4+2], py2 = bb_tile[i*4+3];
      const float lx = fmaxf(gx1, px1), ly = fmaxf(gy1, py1);
      const float rx = fminf(gx2, px2), ry = fminf(gy2, py2);
      const float inter = fmaxf(rx-lx, 0.f) * fmaxf(ry-ly, 0.f);
      const float pa    = fmaxf(px2-px1, 0.f) * fmaxf(py2-py1, 0.f);
      const float iou   = inter / (ga + pa - inter + EPSF);

      __builtin_prefetch(prB + (size_t)(a + NT)*NPRO, 0, 1);        // global_prefetch_b8
      const float sc   = __builtin_nontemporal_load(prB + (size_t)a*NPRO); // read-once gather
      const float iou2 = iou*iou;
      const float al   = sc * (iou2*iou2*iou2);                     // score^1 * iou^6
      ovR[a] = iou;      // regular (RT) stores: keep ws hot in the 192 MB L2
      alR[a] = al;

      const float ax = an_tile[i*2+0], ay = an_tile[i*2+1];
      const float din = fminf(fminf(ax-gx1, ay-gy1), fminf(gx2-ax, gy2-ay));
      const float metric = (din > EPSF) ? al : 0.f;

      // register-resident top-13 (fully unrolled shift-insert; no scratch)
      float cvv = metric; int cii = a;
#pragma unroll
      for (int p = 0; p < TOPK; ++p) {
        if (cvv > tv[p]) { float t = tv[p]; int u = tix[p];
                           tv[p] = cvv; tix[p] = cii; cvv = t; cii = u; }
      }
    }
    __syncthreads();
  }

  // merge 256 per-thread top-13 lists -> exact row top-13
#pragma unroll
  for (int k = 0; k < TOPK; ++k) { cv[tid*TOPK+k] = tv[k]; ci[tid*TOPK+k] = tix[k]; }
  __syncthreads();
  for (int k = 0; k < TOPK; ++k) {
    float bv = -2.f; int bj = -1;
    for (int j = tid; j < NCAND; j += NT) { float v = cv[j]; if (v > bv) { bv = v; bj = j; } }
    rv[tid] = bv; ri[tid] = bj; __syncthreads();
    for (int s = NT >> 1; s > 0; s >>= 1) {
      if (tid < s && rv[tid+s] > rv[tid]) { rv[tid] = rv[tid+s]; ri[tid] = ri[tid+s]; }
      __syncthreads();
    }
    if (tid == 0) {
      int w = ri[0];
      topk_ws[row*TOPK + k] = (w >= 0) ? ci[w] : 0;
      if (w >= 0) cv[w] = -3.f;
    }
    __syncthreads();
  }
}

// =====================================================================
// k2: scatter top-13 hits into per-anchor 64-bit gt masks (NM==64!)
// =====================================================================
__global__ __launch_bounds__(NT) void k2_scatter(
    const int*   __restrict__ topk_ws,
    const float* __restrict__ mask_gt,   // (BS,NM,1)
    const float* __restrict__ gtbb,
    const float* __restrict__ anc,
    unsigned long long* __restrict__ posbits)
{
  int e = blockIdx.x * NT + threadIdx.x;
  if (e >= BS*NM*TOPK) return;
  int row = e / TOPK;
  if (mask_gt[row] <= 0.f) return;       // invalid gt rows contribute nothing
  int b = row / NM, m = row % NM;
  int a = topk_ws[e];
  float gx1 = gtbb[row*4+0], gy1 = gtbb[row*4+1];
  float gx2 = gtbb[row*4+2], gy2 = gtbb[row*4+3];
  float ax = anc[a*2+0], ay = anc[a*2+1];
  float din = fminf(fminf(ax-gx1, ay-gy1), fminf(gx2-ax, gy2-ay));
  if (din > EPSF)
    atomicOr(&posbits[(size_t)b*NA + a], 1ull << m);   // global_atomic_or_b64
}

// =====================================================================
// k3: per-anchor conflict resolution + all gather-type outputs + row stats
// =====================================================================
__global__ __launch_bounds__(NT) void k3_resolve(
    const unsigned long long* __restrict__ posbits,
    const float* __restrict__ ov_ws,
    const float* __restrict__ al_ws,
    const int*   __restrict__ gtpro,
    const int*   __restrict__ gtalp,
    const int*   __restrict__ gtads,   // (BS,NM,5)
    const float* __restrict__ gtbb,
    const float* __restrict__ gtcr,    // (BS,NM,8)
    float* __restrict__ out,
    int*   __restrict__ tgt_ws,
    int*   __restrict__ fg_ws,
    float* __restrict__ alt_ws,
    unsigned* __restrict__ pos_align,
    unsigned* __restrict__ pos_ov)
{
  int p = blockIdx.x * NT + threadIdx.x;
  if (p >= (int)PBA) return;
  int b = p / NA, a = p % NA;

  unsigned long long bits = posbits[p];
  int cnt = __popcll(bits);
  int tgt = 0, fg = 0;
  if (cnt == 1) { tgt = __ffsll(bits) - 1; fg = 1; }
  else if (cnt > 1) {                       // multi-assigned -> iou argmax over m
    float best = -1.f; int bm = 0;
    const float* col = ov_ws + (size_t)b*NM*NA + a;
    for (int m = 0; m < NM; ++m) { float v = col[(size_t)m*NA]; if (v > best) { best = v; bm = m; } }
    tgt = bm; fg = 1;
  }

  const int gi = b*NM + tgt;
  st_nt(&out[OFF_TPRO + p], (float)max(gtpro[gi], 0));
  st_nt(&out[OFF_TALP + p], (float)max(gtalp[gi], 0));
#pragma unroll
  for (int i = 0; i < 5; ++i)
    st_nt(&out[OFF_TADS + (size_t)i*PBA + p], (float)max(gtads[gi*5 + i], 0));
#pragma unroll
  for (int j = 0; j < 4; ++j)
    st_nt(&out[OFF_TBB + (size_t)p*4 + j], gtbb[gi*4 + j]);
#pragma unroll
  for (int j = 0; j < 8; ++j)
    st_nt(&out[OFF_TCR + (size_t)p*8 + j], gtcr[gi*8 + j]);
  st_nt(&out[OFF_FG + p], fg ? 1.f : 0.f);

  tgt_ws[p] = tgt; fg_ws[p] = fg;
  float alt = 0.f;
  if (fg) {
    alt = al_ws[(size_t)gi*NA + a];
    float ovv = ov_ws[(size_t)gi*NA + a];
    // values >= 0 -> float order == uint bit order; monotonic atomic max
    atomicMax(&pos_align[gi], __float_as_uint(alt));
    atomicMax(&pos_ov[gi],    __float_as_uint(ovv));
  }
  alt_ws[p] = alt;
}

// =====================================================================
// k4: coalesced one-hot fills (non-temporal output stream)
// =====================================================================
__global__ __launch_bounds__(NT) void k4_pro(
    const int* __restrict__ tgt_ws, const int* __restrict__ fg_ws,
    const float* __restrict__ alt_ws,
    const unsigned* __restrict__ pos_align, const unsigned* __restrict__ pos_ov,
    const int* __restrict__ gtpro, float* __restrict__ out)
{
  size_t i = (size_t)blockIdx.x * NT + threadIdx.x;
  if (i >= (size_t)NPRO * PBA) return;
  int c = (int)(i % NPRO);
  int p = (int)(i / NPRO);
  float v = 0.f;
  if (fg_ws[p]) {
    int b = p / NA;
    int gi = b*NM + tgt_ws[p];
    if (c == max(gtpro[gi], 0)) {
      float pa = __uint_as_float(pos_align[gi]);
      float po = __uint_as_float(pos_ov[gi]);
      v = alt_ws[p] * po / (pa + EPSF);   // norm
    }
  }
  st_nt(&out[OFF_PRO + i], v);
}

__global__ __launch_bounds__(NT) void k4_alp(
    const int* __restrict__ tgt_ws, const int* __restrict__ fg_ws,
    const int* __restrict__ gtalp, float* __restrict__ out)
{
  size_t i = (size_t)blockIdx.x * NT + threadIdx.x;
  if (i >= (size_t)NALP * PBA) return;
  int c = (int)(i % NALP);
  int p = (int)(i / NALP);
  float v = 0.f;
  if (fg_ws[p]) {
    int b = p / NA;
    if (c == max(gtalp[b*NM + tgt_ws[p]], 0)) v = 1.f;
  }
  st_nt(&out[OFF_ALP + i], v);
}

__global__ __launch_bounds__(NT) void k4_ads(
    const int* __restrict__ tgt_ws, const int* __restrict__ fg_ws,
    const int* __restrict__ gtads, float* __restrict__ out)
{
  size_t i = (size_t)blockIdx.x * NT + threadIdx.x;
  if (i >= (size_t)5 * NADS * PBA) return;
  int c = (int)(i % NADS);
  size_t r = i / NADS;                 // = ii*PBA + p
  int p  = (int)(r % PBA);
  int ii = (int)(r / PBA);
  float v = 0.f;
  if (fg_ws[p]) {
    int b = p / NA;
    if (c == max(gtads[(b*NM + tgt_ws[p])*5 + ii], 0)) v = 1.f;
  }
  st_nt(&out[OFF_ADS + i], v);
}

// =====================================================================
// launch
// =====================================================================
extern "C" void kernel_launch(void* const* d_in, const int* in_sizes, int n_in,
                              void* d_out, int out_size, void* d_ws, size_t ws_size,
                              hipStream_t stream) {
  const float* pd_pro  = (const float*)d_in[0];
  const float* pd_bb   = (const float*)d_in[3];
  const float* anc     = (const float*)d_in[5];
  const int*   gt_pro  = (const int*)  d_in[6];
  const int*   gt_alp  = (const int*)  d_in[7];
  const int*   gt_ads  = (const int*)  d_in[8];
  const float* gt_bb   = (const float*)d_in[9];
  const float* gt_cr   = (const float*)d_in[10];
  const float* mask_gt = (const float*)d_in[11];
  float* out = (float*)d_out;

  // ---- workspace carve-up (all 16B aligned; total ~143 MB, L2-resident hot) ----
  char* w = (char*)d_ws;
  auto take = [&](size_t bytes) { char* p = w; w += (bytes + 15) & ~(size_t)15; return p; };
  const size_t ROWS = (size_t)BS * NM;                 // 2048
  float*              ov_ws   = (float*)take(ROWS * NA * sizeof(float));
  float*              al_ws   = (float*)take(ROWS * NA * sizeof(float));
  unsigned long long* posbits = (unsigned long long*)take(PBA * sizeof(unsigned long long));
  int*                topk_ws = (int*)take(ROWS * TOPK * sizeof(int));
  unsigned*           pos_al  = (unsigned*)take(ROWS * sizeof(unsigned));
  unsigned*           pos_ov  = (unsigned*)take(ROWS * sizeof(unsigned));
  int*                tgt_ws  = (int*)take(PBA * sizeof(int));
  int*                fg_ws   = (int*)take(PBA * sizeof(int));
  float*              alt_ws  = (float*)take(PBA * sizeof(float));
  if ((size_t)(w - (char*)d_ws) > ws_size) return;     // ws too small; bail deterministically

  const int gInit = (int)((PBA + NT - 1) / NT);
  k0_init<<<gInit, NT, 0, stream>>>(posbits, pos_al, pos_ov);

  k1_metrics_topk<<<(int)ROWS, NT, 0, stream>>>(pd_pro, pd_bb, anc, gt_pro, gt_bb,
                                                ov_ws, al_ws, topk_ws);

  const int nScatter = BS * NM * TOPK;
  k2_scatter<<<(nScatter + NT - 1) / NT, NT, 0, stream>>>(topk_ws, mask_gt, gt_bb, anc, posbits);

  k3_resolve<<<gInit, NT, 0, stream>>>(posbits, ov_ws, al_ws, gt_pro, gt_alp, gt_ads,
                                       gt_bb, gt_cr, out, tgt_ws, fg_ws, alt_ws,
                                       pos_al, pos_ov);

  k4_pro<<<(int)((NPRO * PBA + NT - 1) / NT), NT, 0, stream>>>(tgt_ws, fg_ws, alt_ws,
                                                               pos_al, pos_ov, gt_pro, out);
  k4_alp<<<(int)((NALP * PBA + NT - 1) / NT), NT, 0, stream>>>(tgt_ws, fg_ws, gt_alp, out);
  k4_ads<<<(int)((5ull * NADS * PBA + NT - 1) / NT), NT, 0, stream>>>(tgt_ws, fg_ws, gt_ads, out);
}